// QuadrupletLoss_64914135712280
// MI455X (gfx1250) — compile-verified
//
#include <hip/hip_runtime.h>
#include <hip/hip_bf16.h>
#include <math.h>

// ---------------------------------------------------------------------------
// QuadrupletLoss (supervised-contrastive-style) for MI455X / gfx1250.
//
// Key algebraic reduction: the reference's O(N^3) chunked log-sum-exp depends
// on row i only via lab[i]  ->  collapse to NUM_CLASSES=32 per-class masked
// reductions over the NxN similarity matrix. Remaining heavy op is the
// 1024x1024x1024 Gram matrix (2.15 GFLOP, 4MB in / 4MB out, L2-resident),
// done with exact-precision V_WMMA_F32_16X16X4_F32.
//
// Workspace layout (needs 4MB + 2KB):
//   [0, 4MB)        : M[1024][1024] f32  (masked, /TEMP Gram matrix)
//   [4MB, 4MB+2KB)  : partial[32][16] f32 (per-class per-jblock exp-sums)
// ---------------------------------------------------------------------------

#define NCLS   32
#define NN     1024
#define DD     1024
#define TEMP_INV 20.0f          // 1 / 0.05

typedef float v2f __attribute__((ext_vector_type(2)));
typedef float v8f __attribute__((ext_vector_type(8)));

// ---------------------------------------------------------------------------
// Kernel 1: M[j,k] = (lab[j]!=lab[k]) ? dot(a_j, a_k)/TEMP : 0
// 512 threads = 16 waves; block computes a 64x64 tile; each wave one 16x16
// subtile via V_WMMA_F32_16X16X4_F32 (A 16x4, B 4x16, C 16x16 f32).
// LDS tiles strided by 68 floats: b64 reads from 16 rows x 2 cols touch all
// 64 LDS banks exactly once (68 mod 64 = 4).
// ---------------------------------------------------------------------------
__global__ void __launch_bounds__(512)
gram_wmma_kernel(const float* __restrict__ A, const int* __restrict__ labels,
                 float* __restrict__ M) {
  constexpr int LDT = 68;
  __shared__ float At[64 * LDT];
  __shared__ float Bt[64 * LDT];

  const int tid  = threadIdx.x;
  const int wave = tid >> 5;        // 0..15
  const int lane = tid & 31;
  const int lo   = lane & 15;       // N index within tile (and M for A-matrix)
  const int hi   = lane >> 4;       // half-wave select
  const int wm   = wave >> 2;       // 0..3 : row subtile
  const int wn   = wave & 3;        // 0..3 : col subtile
  const int rowB = blockIdx.y * 64;
  const int colB = blockIdx.x * 64;

  v8f acc = {};                     // 16x16 f32 accumulator (8 VGPRs, wave32)

  for (int k0 = 0; k0 < DD; k0 += 64) {
    // Stage A-rows [rowB,rowB+64) and A-rows [colB,colB+64), K-chunk of 64.
    // 2 phases x 512 threads x float4 = 4096 floats per tile.
#pragma unroll
    for (int p = 0; p < 2; ++p) {
      const int idx = tid + p * 512;      // 0..1023
      const int r   = idx >> 4;           // 0..63
      const int c4  = (idx & 15) << 2;    // 0,4,...,60
      const float4 va = *(const float4*)(A + (size_t)(rowB + r) * DD + k0 + c4);
      const float4 vb = *(const float4*)(A + (size_t)(colB + r) * DD + k0 + c4);
      *(float4*)(&At[r * LDT + c4]) = va; // 272B row pitch -> 16B aligned
      *(float4*)(&Bt[r * LDT + c4]) = vb;
    }
    __syncthreads();

    // 16 WMMAs per K-chunk. A-matrix layout: lane(lo)=M row, VGPR v holds
    // K = 2*hi + v. B-matrix mirrors it with lo = N column.
#pragma unroll
    for (int kk = 0; kk < 64; kk += 4) {
      const float* ap = &At[(wm * 16 + lo) * LDT + kk + 2 * hi];
      const float* bp = &Bt[(wn * 16 + lo) * LDT + kk + 2 * hi];
      v2f a, b;
      a[0] = ap[0]; a[1] = ap[1];
      b[0] = bp[0]; b[1] = bp[1];
      acc = __builtin_amdgcn_wmma_f32_16x16x4_f32(
          /*neg_a=*/false, a, /*neg_b=*/false, b,
          /*c_mod=*/(short)0, acc, /*reuse_a=*/false, /*reuse_b=*/false);
    }
    __syncthreads();
  }

  // Epilogue: C/D layout -> VGPR r holds (M = r + 8*hi, N = lo).
  const int col  = colB + wn * 16 + lo;
  const int labc = labels[col];
#pragma unroll
  for (int r = 0; r < 8; ++r) {
    const int row = rowB + wm * 16 + hi * 8 + r;
    const float v = (labels[row] == labc) ? 0.f : acc[r] * TEMP_INV;
    M[(size_t)row * NN + col] = v;
  }
}

// ---------------------------------------------------------------------------
// Kernel 2: per (class c, j-block of 64 rows):
//   partial[c][jb] = sum_{j in block, lab[j]!=c}
//                      sum_{k, lab[k]!=c} exp(M[j,k] - max_{k,lab[k]!=c} M[j,k])
// 256 threads: 4 threads per row, each scans 256 k's. All reductions are in
// fixed order (no float atomics) so results are bit-deterministic.
// ---------------------------------------------------------------------------
__global__ void __launch_bounds__(256)
class_reduce_kernel(const float* __restrict__ M, const int* __restrict__ labels,
                    float* __restrict__ partial) {
  const int c  = blockIdx.x;   // class 0..31
  const int jb = blockIdx.y;   // j-block 0..15

  __shared__ int   labS[NN];
  __shared__ float maxS[64][4];
  __shared__ float sumS[64][4];
  __shared__ float rowS[64];

  const int tid = threadIdx.x;
  for (int i = tid; i < NN; i += 256) labS[i] = labels[i];
  __syncthreads();

  const int jl = tid >> 2;          // 0..63
  const int q  = tid & 3;           // 0..3
  const int j  = jb * 64 + jl;
  const float* Mrow = M + (size_t)j * NN;
  const int k0 = q * 256;

  // pass 1: masked max over this thread's k-quarter
  float mx = -INFINITY;
  for (int k = k0; k < k0 + 256; ++k)
    if (labS[k] != c) mx = fmaxf(mx, Mrow[k]);
  maxS[jl][q] = mx;
  __syncthreads();

  const float rmax = fmaxf(fmaxf(maxS[jl][0], maxS[jl][1]),
                           fmaxf(maxS[jl][2], maxS[jl][3]));

  // pass 2: masked exp-sum against the shared row max
  float s = 0.f;
  if (rmax > -INFINITY) {
    for (int k = k0; k < k0 + 256; ++k)
      if (labS[k] != c) s += __expf(Mrow[k] - rmax);
  }
  sumS[jl][q] = s;
  __syncthreads();

  if (q == 0) {
    const float rs = sumS[jl][0] + sumS[jl][1] + sumS[jl][2] + sumS[jl][3];
    rowS[jl] = (labS[j] != c) ? rs : 0.f;   // gate rows by lab[j]!=c
  }
  __syncthreads();

  if (tid == 0) {
    float t = 0.f;
    for (int i = 0; i < 64; ++i) t += rowS[i];   // fixed order
    partial[c * 16 + jb] = t;
  }
}

// ---------------------------------------------------------------------------
// Kernel 3: one wave. Histogram of labels, closed-form counts_c, final loss:
//   counts_c = (N-n_c)^2 - (sum_c' n^2 - n_c^2)
//   loss = (1/N) * sum_c n_c * [n_c>1] * log(sums_c/counts_c + 1e-12)
// ---------------------------------------------------------------------------
__global__ void __launch_bounds__(32)
finalize_kernel(const int* __restrict__ labels, const float* __restrict__ partial,
                float* __restrict__ out) {
  __shared__ int   hist[NCLS];
  __shared__ float red[NCLS];
  const int c = threadIdx.x;       // 0..31

  int n = 0;
  for (int i = 0; i < NN; ++i) n += (labels[i] == c) ? 1 : 0;
  hist[c] = n;
  __syncthreads();

  float sumsq = 0.f;
  for (int i = 0; i < NCLS; ++i) { const float v = (float)hist[i]; sumsq += v * v; }

  const float nc     = (float)n;
  const float other  = (float)NN - nc;
  const float counts = other * other - (sumsq - nc * nc);

  float s = 0.f;
  for (int j = 0; j < 16; ++j) s += partial[c * 16 + j];

  const float x = s / counts;
  red[c] = (n > 1) ? nc * logf(x + 1e-12f) : 0.f;
  __syncthreads();

  if (c == 0) {
    float t = 0.f;
    for (int i = 0; i < NCLS; ++i) t += red[i];
    out[0] = t / (float)NN;
  }
}

// ---------------------------------------------------------------------------
extern "C" void kernel_launch(void* const* d_in, const int* in_sizes, int n_in,
                              void* d_out, int out_size, void* d_ws, size_t ws_size,
                              hipStream_t stream) {
  const float* anchor = (const float*)d_in[0];
  // d_in[1] (target) is unused by the reference loss.
  const int* labels = (const int*)d_in[2];

  float* Mbuf    = (float*)d_ws;
  float* partial = (float*)((char*)d_ws + (size_t)NN * NN * sizeof(float));
  float* out     = (float*)d_out;

  dim3 g1(NN / 64, NN / 64);   // 16 x 16 blocks
  gram_wmma_kernel<<<g1, 512, 0, stream>>>(anchor, labels, Mbuf);

  dim3 g2(NCLS, NN / 64);      // 32 classes x 16 j-blocks
  class_reduce_kernel<<<g2, 256, 0, stream>>>(Mbuf, labels, partial);

  finalize_kernel<<<1, 32, 0, stream>>>(labels, partial, out);
}